// LocalCrossLinearTrf_27333171872134
// MI455X (gfx1250) — compile-verified
//
#include <hip/hip_runtime.h>
#include <cstdint>

// ---------------------------------------------------------------------------
// LocalCrossLinearTrf on MI455X (gfx1250, wave32)
//
// Roofline: ~296 MB of fp32 traffic -> ~13 us HBM floor @ 23.3 TB/s;
// ~1.7e9 VALU lane-ops -> comparable VALU time. Memory path choices:
//   * trf (192 MB, dominant stream): double-buffered per-thread rows staged
//     through LDS with gfx1250 async global->LDS loads + s_wait_asynccnt.
//   * x (16 MB): data-dependent trilinear gathers; fits entirely in the
//     192 MB L2, lanes = consecutive w-voxels -> 32B-strided addresses.
//   * No WMMA: the contraction is a per-voxel Hadamard+reduce over K=8 with
//     both operands voxel-indexed; matrix units cannot express it.
// ---------------------------------------------------------------------------

#define NB    2
#define DIMV  64
#define VOL   (DIMV * DIMV * DIMV)
#define FI    8
#define FO    8
#define TPB   256
#define ROWF  28   // padded LDS row stride in floats (24 used) -> <=2-way bank conflict

#if defined(__AMDGCN__) && __has_builtin(__builtin_amdgcn_global_load_async_to_lds_b128)
#define USE_ASYNC 1
#else
#define USE_ASYNC 0
#endif

#if defined(__AMDGCN__) && __has_builtin(__builtin_amdgcn_s_wait_asynccnt)
#define WAIT_ASYNC(n) __builtin_amdgcn_s_wait_asynccnt(n)
#elif USE_ASYNC
#define WAIT_ASYNC(n) asm volatile("s_wait_asynccnt %0" ::"i"(n) : "memory")
#else
#define WAIT_ASYNC(n) do {} while (0)
#endif

#if USE_ASYNC
typedef int v4i_t __attribute__((ext_vector_type(4)));
typedef __attribute__((address_space(1))) v4i_t glb_v4i;  // global int4 (prints as __device__ int4*)
typedef __attribute__((address_space(3))) v4i_t lds_v4i;  // LDS int4
#endif

__device__ __forceinline__ void stage_row(const float* __restrict__ g, float* l) {
#if USE_ASYNC
  #pragma unroll
  for (int c = 0; c < 6; ++c) {
    __builtin_amdgcn_global_load_async_to_lds_b128(
        (glb_v4i*)(g + 4 * c),
        (lds_v4i*)(l + 4 * c),
        /*offset=*/0, /*cpol=*/0);
  }
#else
  #pragma unroll
  for (int c = 0; c < 6; ++c)
    reinterpret_cast<float4*>(l)[c] = reinterpret_cast<const float4*>(g)[c];
#endif
}

__global__ __launch_bounds__(TPB) void lclt_kernel(
    const float* __restrict__ x,     // [B, D, H, W, FI]
    const float* __restrict__ mult,  // [D, H, W, FI, FO]
    const float* __restrict__ trf,   // [D, H, W, FI, FO, 3]
    const float* __restrict__ bias,  // [D, H, W, FO]
    float* __restrict__ out)         // [B, D, H, W, FO]
{
  __shared__ float sbuf[2 * TPB * ROWF];  // 57344 B: double-buffered trf rows

  const int t   = threadIdx.x;
  const int gid = blockIdx.x * TPB + t;      // 0 .. NB*VOL-1
  const int v   = gid & (VOL - 1);           // voxel id
  const int b   = gid >> 18;                 // batch id
  const int iw  = v & 63;
  const int ih  = (v >> 6) & 63;
  const int id  = (v >> 12) & 63;
  const float fd = (float)id, fh = (float)ih, fw = (float)iw;

  const float* trf_v  = trf  + (size_t)v * (FI * FO * 3);
  const float* mult_v = mult + (size_t)v * (FI * FO);
  const float* bias_v = bias + (size_t)v * FO;
  const float* xb     = x    + (size_t)b * ((size_t)VOL * FI);

  __builtin_prefetch(mult_v, 0, 0);   // global_prefetch_b8
  __builtin_prefetch(bias_v, 0, 0);

  float* row0 = &sbuf[t * ROWF];
  float* row1 = &sbuf[TPB * ROWF + t * ROWF];

  float acc[FO];
  #pragma unroll
  for (int j = 0; j < FO; ++j) acc[j] = 0.0f;

  stage_row(trf_v, row0);  // prime the pipeline with i=0

  #pragma unroll
  for (int i = 0; i < FI; ++i) {
    float* cur = (i & 1) ? row1 : row0;
    float* nxt = (i & 1) ? row0 : row1;
    if (i + 1 < FI) {
      stage_row(trf_v + (i + 1) * (FO * 3), nxt);  // prefetch next feature's rows
      WAIT_ASYNC(6);  // 6 newer ops outstanding -> tile i is complete (in-order)
    } else {
      WAIT_ASYNC(0);
    }

    float r[FO * 3];
    #pragma unroll
    for (int c = 0; c < 6; ++c)
      *reinterpret_cast<float4*>(&r[4 * c]) = reinterpret_cast<const float4*>(cur)[c];

    float m[FO];
    *reinterpret_cast<float4*>(&m[0]) = reinterpret_cast<const float4*>(mult_v + i * FO)[0];
    *reinterpret_cast<float4*>(&m[4]) = reinterpret_cast<const float4*>(mult_v + i * FO)[1];

    #pragma unroll
    for (int j = 0; j < FO; ++j) {
      // sampling location = voxel + displacement, clamped to [0, 63]
      const float cd = fminf(fmaxf(fd + r[3 * j + 0], 0.0f), 63.0f);
      const float ch = fminf(fmaxf(fh + r[3 * j + 1], 0.0f), 63.0f);
      const float cw = fminf(fmaxf(fw + r[3 * j + 2], 0.0f), 63.0f);
      const float d0f = floorf(cd), h0f = floorf(ch), w0f = floorf(cw);
      const float d1f = fminf(d0f + 1.0f, 63.0f);
      const float h1f = fminf(h0f + 1.0f, 63.0f);
      const float w1f = fminf(w0f + 1.0f, 63.0f);
      // bit==0 corner (l0) gets weight d1 = l1 - cl ; bit==1 corner (l1) gets 1 - d1
      const float wdl = d1f - cd, wdh = 1.0f - wdl;
      const float whl = h1f - ch, whh = 1.0f - whl;
      const float wwl = w1f - cw, wwh = 1.0f - wwl;
      const int d0 = (int)d0f, d1 = (int)d1f;
      const int h0 = (int)h0f, h1 = (int)h1f;
      const int w0 = (int)w0f, w1 = (int)w1f;
      // row bases: ((d*64 + h)*64)*8 elements; element = base + w*8 + i
      const int r00 = ((d0 << 6) + h0) << 9;
      const int r01 = ((d0 << 6) + h1) << 9;
      const int r10 = ((d1 << 6) + h0) << 9;
      const int r11 = ((d1 << 6) + h1) << 9;
      const int cw0 = (w0 << 3) + i;
      const int cw1 = (w1 << 3) + i;

      const float v000 = xb[r00 + cw0], v001 = xb[r00 + cw1];
      const float v010 = xb[r01 + cw0], v011 = xb[r01 + cw1];
      const float v100 = xb[r10 + cw0], v101 = xb[r10 + cw1];
      const float v110 = xb[r11 + cw0], v111 = xb[r11 + cw1];

      const float p00 = wdl * whl, p01 = wdl * whh;
      const float p10 = wdh * whl, p11 = wdh * whh;
      float s;
      s = (p00 * wwl) * v000;
      s = fmaf(p00 * wwh, v001, s);
      s = fmaf(p01 * wwl, v010, s);
      s = fmaf(p01 * wwh, v011, s);
      s = fmaf(p10 * wwl, v100, s);
      s = fmaf(p10 * wwh, v101, s);
      s = fmaf(p11 * wwl, v110, s);
      s = fmaf(p11 * wwh, v111, s);

      acc[j] = fmaf(s, m[j], acc[j]);
    }
  }

  float bs[FO];
  *reinterpret_cast<float4*>(&bs[0]) = reinterpret_cast<const float4*>(bias_v)[0];
  *reinterpret_cast<float4*>(&bs[4]) = reinterpret_cast<const float4*>(bias_v)[1];

  float o[FO];
  #pragma unroll
  for (int j = 0; j < FO; ++j) o[j] = fmaf(8.0f, bs[j], acc[j]);  // + F_IN * bias

  float* outp = out + (size_t)gid * FO;
  reinterpret_cast<float4*>(outp)[0] = *reinterpret_cast<float4*>(&o[0]);
  reinterpret_cast<float4*>(outp)[1] = *reinterpret_cast<float4*>(&o[4]);
}

extern "C" void kernel_launch(void* const* d_in, const int* in_sizes, int n_in,
                              void* d_out, int out_size, void* d_ws, size_t ws_size,
                              hipStream_t stream) {
  const float* x    = (const float*)d_in[0];  // [B,D,H,W,FI]
  const float* mult = (const float*)d_in[1];  // [D,H,W,FI,FO]
  const float* trf  = (const float*)d_in[2];  // [D,H,W,FI,FO,3]
  const float* bias = (const float*)d_in[3];  // [D,H,W,FO]
  float* out = (float*)d_out;                 // [B,D,H,W,FO]

  const int threads = NB * VOL;               // one thread per (batch, voxel)
  const int blocks  = threads / TPB;          // 2048
  lclt_kernel<<<blocks, TPB, 0, stream>>>(x, mult, trf, bias, out);
}